// GCN_21294447853629
// MI455X (gfx1250) — compile-verified
//
#include <hip/hip_runtime.h>

typedef __attribute__((ext_vector_type(2))) float v2f;
typedef __attribute__((ext_vector_type(8))) float v8f;

#define NCH 128

// ---------------------------------------------------------------- utilities
__global__ void fill_kernel(float* __restrict__ p, float v, size_t n) {
    size_t i = (size_t)blockIdx.x * blockDim.x + threadIdx.x;
    if (i < n) p[i] = v;
}

__global__ void degree_kernel(const int* __restrict__ col, float* __restrict__ deg, int E) {
    int e = blockIdx.x * blockDim.x + threadIdx.x;
    if (e < E) atomicAdd(&deg[col[e]], 1.0f);
}

__global__ void rsqrt_kernel(const float* __restrict__ deg, float* __restrict__ dinv, int n) {
    int i = blockIdx.x * blockDim.x + threadIdx.x;
    if (i < n) {
        float d = deg[i];
        dinv[i] = (d > 0.0f) ? rsqrtf(d) : 0.0f;
    }
}

// ------------------------------------------------- fp32 WMMA GEMM: H = X @ W
// X: [nrows,128], W: [128,128] row-major, H: [nrows,128]
// LDS holds W K-interleaved: Wlds[(k>>2)*512 + n*4 + (k&3)] = W[k][n]
// so each B fragment {W[k0+kh][n], W[k0+kh+1][n]} is one contiguous b64 load.
__global__ __launch_bounds__(256) void gemm128_wmma(const float* __restrict__ X,
                                                    const float* __restrict__ W,
                                                    float* __restrict__ H, int nrows) {
    __shared__ float Wlds[32 * 512];  // 16384 floats = 64 KB
    const int tid = threadIdx.x;
    // cooperative repack: unit u = (kgroup, n); 4 column dwords -> 1 b128 store
#pragma unroll
    for (int i = 0; i < 16; ++i) {
        int u  = i * 256 + tid;  // 0..4095
        int kg = u >> 7;         // k group of 4: 0..31
        int n  = u & 127;
        float4 v;
        v.x = W[(kg * 4 + 0) * NCH + n];
        v.y = W[(kg * 4 + 1) * NCH + n];
        v.z = W[(kg * 4 + 2) * NCH + n];
        v.w = W[(kg * 4 + 3) * NCH + n];
        *(float4*)(&Wlds[kg * 512 + n * 4]) = v;
    }
    __syncthreads();

    const int lane  = tid & 31;
    const int wave  = tid >> 5;
    const int rowBase = blockIdx.x * 128 + wave * 16;
    const int mlane = lane & 15;
    const int khalf = (lane >> 4) * 2;  // 0 for lanes 0-15, 2 for lanes 16-31

    int arow = rowBase + mlane;
    if (arow >= nrows) arow = nrows - 1;  // clamp (WMMA needs EXEC all-ones)
    const float* xrow = X + (size_t)arow * NCH;

    v8f acc[8];
#pragma unroll
    for (int t = 0; t < 8; ++t) acc[t] = (v8f){0, 0, 0, 0, 0, 0, 0, 0};

    // per-lane LDS base: column (t*16+mlane)*4 + khalf, kgroup via loop offset
    const float* wcol = &Wlds[mlane * 4 + khalf];

    for (int k0 = 0; k0 < NCH; k0 += 4) {
        // A fragment 16x4: lane holds X[row][k0+khalf .. +1]
        const float2 av = *(const float2*)(xrow + k0 + khalf);
        v2f a; a.x = av.x; a.y = av.y;
        const float* wk = wcol + (k0 >> 2) * 512;
#pragma unroll
        for (int t = 0; t < 8; ++t) {
            const float2 bv = *(const float2*)(wk + t * 64);  // single ds_load_b64
            v2f b; b.x = bv.x; b.y = bv.y;
            acc[t] = __builtin_amdgcn_wmma_f32_16x16x4_f32(
                false, a, false, b, (short)0, acc[t], false, false);
        }
    }

    // C/D layout: VGPR i -> M=i (lanes 0-15) / M=i+8 (lanes 16-31), N=lane&15
    const int rowOff = (lane >> 4) * 8;
#pragma unroll
    for (int t = 0; t < 8; ++t) {
#pragma unroll
        for (int i = 0; i < 8; ++i) {
            int row = rowBase + rowOff + i;
            if (row < nrows) H[(size_t)row * NCH + t * 16 + mlane] = acc[t][i];
        }
    }
}

// ------------------------------------- edge scatter: agg[c] += norm * H[r]
// one wave per edge (incl. implicit self-loops at e >= E); 32 lanes x float4
__global__ __launch_bounds__(256) void scatter_kernel(const float* __restrict__ H,
                                                      const int* __restrict__ row,
                                                      const int* __restrict__ col,
                                                      const float* __restrict__ dinv,
                                                      float* __restrict__ agg,
                                                      int nEdges, int nNodes) {
    const int lane = threadIdx.x & 31;
    const int wv   = threadIdx.x >> 5;
    const long long e     = (long long)blockIdx.x * 8 + wv;
    const long long total = (long long)nEdges + nNodes;
    if (e >= total) return;
    int r, c;
    if (e < nEdges) { r = row[e]; c = col[e]; }
    else            { r = c = (int)(e - nEdges); }
    const float nrm = dinv[r] * dinv[c];
    const float4 v  = *(const float4*)(H + (size_t)r * NCH + lane * 4);
    float* dst = agg + (size_t)c * NCH + lane * 4;
    atomicAdd(dst + 0, nrm * v.x);
    atomicAdd(dst + 1, nrm * v.y);
    atomicAdd(dst + 2, nrm * v.z);
    atomicAdd(dst + 3, nrm * v.w);
}

__global__ void bias_relu_kernel(const float* __restrict__ agg, const float* __restrict__ bias,
                                 float* __restrict__ out, size_t total) {
    size_t i = (size_t)blockIdx.x * blockDim.x + threadIdx.x;
    if (i < total) {
        float v = agg[i] + bias[i & (NCH - 1)];
        out[i] = v > 0.0f ? v : 0.0f;
    }
}

// ----------------------- mean-pool (batch is sorted: register-accumulate)
#define NODES_PER_BLOCK 1024
__global__ __launch_bounds__(128) void pool_kernel(const float* __restrict__ H,
                                                   const int* __restrict__ batch,
                                                   float* __restrict__ sums,
                                                   float* __restrict__ cnts, int nNodes) {
    const int ch    = threadIdx.x;  // 0..127
    const int start = blockIdx.x * NODES_PER_BLOCK;
    int end = start + NODES_PER_BLOCK;
    if (end > nNodes) end = nNodes;
    float acc = 0.0f, cnt = 0.0f;
    int cur = -1;
    for (int n = start; n < end; ++n) {
        int g = batch[n];  // uniform across block
        if (g != cur) {
            if (cur >= 0) {
                atomicAdd(&sums[cur * NCH + ch], acc);
                if (ch == 0) atomicAdd(&cnts[cur], cnt);
            }
            acc = 0.0f; cnt = 0.0f; cur = g;
        }
        acc += H[(size_t)n * NCH + ch];
        cnt += 1.0f;
    }
    if (cur >= 0) {
        atomicAdd(&sums[cur * NCH + ch], acc);
        if (ch == 0) atomicAdd(&cnts[cur], cnt);
    }
}

// --------------------------- head: out[g,o] = mean(sums[g]) @ Wlin + blin
__global__ __launch_bounds__(128) void head_kernel(const float* __restrict__ sums,
                                                   const float* __restrict__ cnts,
                                                   const float* __restrict__ Wlin,
                                                   const float* __restrict__ blin,
                                                   float* __restrict__ out) {
    int t = threadIdx.x;        // 0..127
    int g = t >> 1, o = t & 1;  // 64 graphs x 2 outputs
    float c = cnts[g];
    if (c < 1.0f) c = 1.0f;
    float inv = 1.0f / c;
    float s = 0.0f;
    for (int k = 0; k < NCH; ++k) s += sums[g * NCH + k] * inv * Wlin[k * 2 + o];
    out[g * 2 + o] = s + blin[o];
}

// --------------------------------------------------------------- launcher
extern "C" void kernel_launch(void* const* d_in, const int* in_sizes, int n_in,
                              void* d_out, int out_size, void* d_ws, size_t ws_size,
                              hipStream_t stream) {
    const float* x    = (const float*)d_in[0];
    const int*   ei   = (const int*)d_in[1];
    const int*   batch= (const int*)d_in[2];
    const float* W1   = (const float*)d_in[3];
    const float* b1   = (const float*)d_in[4];
    const float* W2   = (const float*)d_in[5];
    const float* b2   = (const float*)d_in[6];
    const float* Wlin = (const float*)d_in[7];
    const float* blin = (const float*)d_in[8];
    float*       out  = (float*)d_out;

    const int N = in_sizes[0] / NCH;   // 100000
    const int E = in_sizes[1] / 2;     // 1600000
    const int G = 64;
    const int* row = ei;
    const int* col = ei + E;

    const size_t NC = (size_t)N * NCH;
    float* bufA = (float*)d_ws;
    float* bufB = bufA + NC;
    float* deg  = bufB + NC;
    float* dinv = deg + N;
    float* sums = dinv + N;
    float* cnts = sums + (size_t)G * NCH;

    const int gemmGrid = (N + 127) / 128;
    const long long totEdges = (long long)E + N;
    const int scatGrid = (int)((totEdges + 7) / 8);
    const int elwGrid  = (int)((NC + 255) / 256);

    // degrees & normalization
    fill_kernel<<<(N + 255) / 256, 256, 0, stream>>>(deg, 1.0f, (size_t)N);  // self-loop
    degree_kernel<<<(E + 255) / 256, 256, 0, stream>>>(col, deg, E);
    rsqrt_kernel<<<(N + 255) / 256, 256, 0, stream>>>(deg, dinv, N);

    // layer 1
    gemm128_wmma<<<gemmGrid, 256, 0, stream>>>(x, W1, bufA, N);
    fill_kernel<<<elwGrid, 256, 0, stream>>>(bufB, 0.0f, NC);
    scatter_kernel<<<scatGrid, 256, 0, stream>>>(bufA, row, col, dinv, bufB, E, N);
    bias_relu_kernel<<<elwGrid, 256, 0, stream>>>(bufB, b1, bufA, NC);

    // layer 2
    gemm128_wmma<<<gemmGrid, 256, 0, stream>>>(bufA, W2, bufB, N);
    fill_kernel<<<elwGrid, 256, 0, stream>>>(bufA, 0.0f, NC);
    scatter_kernel<<<scatGrid, 256, 0, stream>>>(bufB, row, col, dinv, bufA, E, N);
    bias_relu_kernel<<<elwGrid, 256, 0, stream>>>(bufA, b2, bufB, NC);

    // pool + head
    fill_kernel<<<((size_t)G * NCH + 255) / 256, 256, 0, stream>>>(sums, 0.0f, (size_t)G * NCH);
    fill_kernel<<<1, 256, 0, stream>>>(cnts, 0.0f, (size_t)G);
    pool_kernel<<<(N + NODES_PER_BLOCK - 1) / NODES_PER_BLOCK, 128, 0, stream>>>(bufB, batch, sums, cnts, N);
    head_kernel<<<1, 128, 0, stream>>>(sums, cnts, Wlin, blin, out);
}